// TrainableMovingAverage1d_12103217840658
// MI455X (gfx1250) — compile-verified
//
#include <hip/hip_runtime.h>

typedef __attribute__((ext_vector_type(2))) float v2f;
typedef __attribute__((ext_vector_type(8))) float v8f;

#define ROWS   4096
#define COLS   8192
#define KS     25
#define GCOLS  (COLS - KS)      /* 8167 valid g columns */
#define TPW    8                /* 16-col tiles per wave (one 16x128 strip) */
#define WPB    8                /* waves per block (256 threads, wave32)    */

// g[r,c] = sum_{i=0..24} x[r,c+i] * w[i]
// out[:, 25+c] = g[:, c]  (c < 8167);  out[:, 0:25] = g[:, 0:25]
__global__ __launch_bounds__(256) void tma1d_wmma_kernel(
    const float* __restrict__ x, const float* __restrict__ w,
    float* __restrict__ out) {
  const int lane = threadIdx.x & 31;
  const int lr   = lane & 15;     // row (A) / column (B,C,D) within tile
  const int hi   = lane >> 4;     // lane-half selects K/M halves
  const int wid  = blockIdx.x * WPB + (threadIdx.x >> 5);

  const int rowTile = wid >> 6;           // 0..255  (4096/16 row tiles)
  const int run     = wid & 63;           // 0..63   (column strips)
  const int r0      = rowTile << 4;
  int       c0      = run * (TPW * 16);   // strip base g-column

  // ---- B: Toeplitz of w, 10 chunks of K=4. VGPR0 holds K=2*hi, VGPR1 K=2*hi+1.
  v2f b[10];
#pragma unroll
  for (int q = 0; q < 10; ++q) {
    const int i0 = 4 * q + 2 * hi - lr;   // tap index for B[2*hi][lr]
    const int i1 = i0 + 1;                // tap index for B[2*hi+1][lr]
    b[q][0] = (i0 >= 0 && i0 < KS) ? w[i0] : 0.0f;
    b[q][1] = (i1 >= 0 && i1 < KS) ? w[i1] : 0.0f;
  }

  const float* xrow = x + (size_t)(r0 + lr) * COLS;
  auto loadA = [&](int colbase) -> v2f {     // A chunk: float2 at K = 2*hi
    int col = colbase + 2 * hi;
    if (col > COLS - 2) col = COLS - 2;      // clamp (dead lanes at right edge)
    return *(const v2f*)(xrow + col);
  };

  // Rolling A buffer: chunk q covers x columns [c0+4q, c0+4q+4)
  v2f a[10];
#pragma unroll
  for (int q = 0; q < 6; ++q) a[q] = loadA(c0 + 4 * q);

  float* orow = out + (size_t)(r0 + 8 * hi) * COLS;

  for (int t = 0; t < TPW; ++t) {
    a[6] = loadA(c0 + 24);
    a[7] = loadA(c0 + 28);
    a[8] = loadA(c0 + 32);
    a[9] = loadA(c0 + 36);

    v8f c = {};
#if __has_builtin(__builtin_amdgcn_wmma_f32_16x16x4_f32)
#pragma unroll
    for (int q = 0; q < 10; ++q) {
      // D = A(16x4) * B(4x16) + C   (exact fp32 path, chained accumulate)
      c = __builtin_amdgcn_wmma_f32_16x16x4_f32(
          false, a[q], false, b[q], (short)0, c, false, false);
    }
#else
    // Fallback (compile-safety only): direct per-lane FMA for rows v+8*hi.
    {
      const int gc = c0 + lr;
#pragma unroll
      for (int v = 0; v < 8; ++v) {
        const float* xr = x + (size_t)(r0 + 8 * hi + v) * COLS;
        float s = 0.0f;
        int cc = (gc > GCOLS - 1) ? (GCOLS - 1) : gc;
        for (int i = 0; i < KS; ++i) s = fmaf(xr[cc + i], w[i], s);
        c[v] = s;
      }
    }
#endif

    // Store: lane writes rows r0+8*hi+v, column gc (shifted by 25; head dup).
    const int gc = c0 + lr;
    if (gc < GCOLS) {
#pragma unroll
      for (int v = 0; v < 8; ++v)
        __builtin_nontemporal_store(c[v], &orow[(size_t)v * COLS + gc + KS]);
    }
    if (gc < KS) {
#pragma unroll
      for (int v = 0; v < 8; ++v)
        __builtin_nontemporal_store(c[v], &orow[(size_t)v * COLS + gc]);
    }

    // Roll: next tile's chunk q == this tile's chunk q+4
#pragma unroll
    for (int q = 0; q < 6; ++q) a[q] = a[q + 4];
    c0 += 16;
  }
}

extern "C" void kernel_launch(void* const* d_in, const int* in_sizes, int n_in,
                              void* d_out, int out_size, void* d_ws, size_t ws_size,
                              hipStream_t stream) {
  const float* x = (const float*)d_in[0];   // (4096, 8192) fp32
  const float* w = (const float*)d_in[1];   // (25, 1) fp32
  float* out = (float*)d_out;               // (4096, 8192) fp32

  // 256 row tiles * 64 strips = 16384 waves; 8 waves/block -> 2048 blocks
  dim3 grid(2048), block(256);
  hipLaunchKernelGGL(tma1d_wmma_kernel, grid, block, 0, stream, x, w, out);
}